// GCN_53498112639137
// MI455X (gfx1250) — compile-verified
//
#include <hip/hip_runtime.h>

#define NN 50000
#define NE 400000
#define NG 1024
#define EPSV 1e-5f

typedef float v2f __attribute__((ext_vector_type(2)));
typedef float v8f __attribute__((ext_vector_type(8)));

// ---------------- degree kernels ----------------
__global__ void gcn_init_deg(float* __restrict__ deg, int n) {
    int i = blockIdx.x * blockDim.x + threadIdx.x;
    if (i < n) deg[i] = 1.0f;   // self-loop included
}

__global__ void gcn_deg_count(const int* __restrict__ ei, float* __restrict__ deg, int E) {
    int e = blockIdx.x * blockDim.x + threadIdx.x;
    if (e < E) atomicAdd(&deg[ei[E + e]], 1.0f);   // dst row of edge_index
}

__global__ void gcn_rsqrt_inplace(float* __restrict__ d, int n) {
    int i = blockIdx.x * blockDim.x + threadIdx.x;
    if (i < n) d[i] = rsqrtf(d[i]);
}

// 8 K-steps of dual WMMA on one staged 32-deep chunk.
// As: [64][34] row-major (m, k); Bs: [64][34] transposed (n, k); even stride
// => every fragment is one aligned ds_load_b64.
__device__ __forceinline__ void wmma_chunk(const float (*As)[34], const float (*Bs)[34],
                                           int wm, int wn, int mr, int kh,
                                           v8f& acc0, v8f& acc1) {
    #pragma unroll
    for (int ks = 0; ks < 8; ++ks) {
        const int k0 = ks * 4 + 2 * kh;     // lane half picks K pair
        v2f a  = *(const v2f*)&As[wm * 16 + mr][k0];
        v2f b0 = *(const v2f*)&Bs[wn * 32 + mr][k0];
        v2f b1 = *(const v2f*)&Bs[wn * 32 + 16 + mr][k0];
        acc0 = __builtin_amdgcn_wmma_f32_16x16x4_f32(
            false, a, false, b0, (short)0, acc0, false, false);
        acc1 = __builtin_amdgcn_wmma_f32_16x16x4_f32(
            false, a, false, b1, (short)0, acc1, false, false);
    }
}

// ---------------- WMMA fp32 GEMM: C[M,N] = A[M,K] @ W[K,N] (+epilogue) ----
// Requirements: N % 64 == 0 (all call sites satisfy). M, K arbitrary.
// Block tile 64x64, BK=32; 8 waves, each wave = 16x32 (two 16x16 accumulators
// sharing one A fragment). mode 0: raw store; 1: +bias,relu; 2: +bias,relu,bn
__global__ __launch_bounds__(256)
void gcn_wmma_gemm(const float* __restrict__ A, const float* __restrict__ W,
                   float* __restrict__ C, int M, int K, int N,
                   const float* __restrict__ bias,
                   const float* __restrict__ bng, const float* __restrict__ bnb,
                   const float* __restrict__ bnm, const float* __restrict__ bnv,
                   int mode) {
    __shared__ __align__(16) float As[64][34];   // (m, k) even stride: b64 frags
    __shared__ __align__(16) float Bs[64][34];   // (n, k) transposed

    const int tid  = threadIdx.x;
    const int lane = tid & 31;
    const int wid  = tid >> 5;     // 8 waves
    const int wm   = wid & 3;      // wave M tile (0..3) -> 16 rows
    const int wn   = wid >> 2;     // wave N supertile (0..1) -> 32 cols
    const int kh   = lane >> 4;    // lane half selects K pair
    const int mr   = lane & 15;    // row (A) / col (B,C) within 16-tile

    const int mBase = blockIdx.y * 64;
    const int nBase = blockIdx.x * 64;

    // fixed per-thread staging coordinates
    const int ar = tid >> 5;       // A row 0..7 (+8*i)
    const int ac = tid & 31;       // A col (k within chunk)
    const int br = tid >> 6;       // B k-subrow 0..3 (+4*i)
    const int bc = tid & 63;       // B col (n)

    // hoisted row pointers (64-bit muls once, not per chunk)
    const float* aPtr[8];
    bool aOk[8];
    #pragma unroll
    for (int i = 0; i < 8; ++i) {
        int m = mBase + i * 8 + ar;
        aOk[i]  = (m < M);
        aPtr[i] = A + (size_t)min(m, M - 1) * K;
    }
    const float* bPtr[8];
    #pragma unroll
    for (int i = 0; i < 8; ++i)
        bPtr[i] = W + (size_t)(i * 4 + br) * N + nBase + bc;

    v8f acc0 = {}, acc1 = {};
    const int  kMain = K & ~31;
    const bool mFull = (mBase + 64 <= M);   // block-uniform

    // ---- main loop: no bounds checks at all ----
    for (int kk = 0; kk < kMain; kk += 32) {
        const size_t koff = (size_t)kk * N;   // uniform scalar
        if (mFull) {
            #pragma unroll
            for (int i = 0; i < 8; ++i)
                As[i * 8 + ar][ac] = aPtr[i][kk + ac];
        } else {
            #pragma unroll
            for (int i = 0; i < 8; ++i) {
                float v = aPtr[i][kk + ac];          // clamped row: always safe
                As[i * 8 + ar][ac] = aOk[i] ? v : 0.0f;
            }
        }
        #pragma unroll
        for (int i = 0; i < 8; ++i)
            Bs[bc][i * 4 + br] = bPtr[i][koff];
        __syncthreads();
        wmma_chunk(As, Bs, wm, wn, mr, kh, acc0, acc1);
        __syncthreads();
    }

    // ---- tail chunk (only K=373 path), fully guarded ----
    if (kMain < K) {
        #pragma unroll
        for (int i = 0; i < 8; ++i) {
            int k = kMain + ac;
            float v = aPtr[i][min(k, K - 1)];
            As[i * 8 + ar][ac] = (aOk[i] && k < K) ? v : 0.0f;
        }
        #pragma unroll
        for (int i = 0; i < 8; ++i) {
            int k = kMain + i * 4 + br;
            float v = W[(size_t)min(k, K - 1) * N + nBase + bc];
            Bs[bc][i * 4 + br] = (k < K) ? v : 0.0f;
        }
        __syncthreads();
        wmma_chunk(As, Bs, wm, wn, mr, kh, acc0, acc1);
        __syncthreads();
    }

    // ---- store + fused epilogue (consecutive rows: one base, += N) ----
    const int rowBase = mBase + wm * 16 + 8 * kh;
    const int col0    = nBase + wn * 32 + mr;
    size_t off = (size_t)rowBase * N + col0;
    #pragma unroll
    for (int r = 0; r < 8; ++r) {
        if (rowBase + r < M) {
            float y0 = acc0[r];
            float y1 = acc1[r];
            if (mode >= 1) {
                y0 += bias[col0];       y0 = fmaxf(y0, 0.0f);
                y1 += bias[col0 + 16];  y1 = fmaxf(y1, 0.0f);
            }
            if (mode == 2) {
                int c1 = col0 + 16;
                y0 = (y0 - bnm[col0]) * (bng[col0] * rsqrtf(bnv[col0] + EPSV)) + bnb[col0];
                y1 = (y1 - bnm[c1])   * (bng[c1]   * rsqrtf(bnv[c1]   + EPSV)) + bnb[c1];
            }
            C[off]      = y0;
            C[off + 16] = y1;
        }
        off += N;
    }
}

// ------ edge aggregation: agg[dst] += h[src] * dis[src]*dis[dst] ----------
// C4 = C/4 is a power of two; c4shift = log2(C4). One thread = 4 channels.
__global__ void gcn_aggregate(const float* __restrict__ h, const int* __restrict__ ei,
                              const float* __restrict__ dis, float* __restrict__ agg,
                              int c4shift, int E, int C) {
    int idx = blockIdx.x * blockDim.x + threadIdx.x;
    int e  = idx >> c4shift;
    int cg = idx & ((1 << c4shift) - 1);
    if (e >= E) return;
    int s = ei[e], d = ei[E + e];
    float nrm = dis[s] * dis[d];
    const float4 hv = *reinterpret_cast<const float4*>(h + (size_t)s * C + cg * 4);
    float* o = agg + (size_t)d * C + cg * 4;
    atomicAdd(o + 0, hv.x * nrm);
    atomicAdd(o + 1, hv.y * nrm);
    atomicAdd(o + 2, hv.z * nrm);
    atomicAdd(o + 3, hv.w * nrm);
}

// ------ fused: y = bn(relu(agg + h*dis^2 + b)), in place over agg ---------
__global__ void gcn_epilogue(float* __restrict__ agg, const float* __restrict__ h,
                             const float* __restrict__ dis, const float* __restrict__ b,
                             const float* __restrict__ g, const float* __restrict__ be,
                             const float* __restrict__ m, const float* __restrict__ v,
                             int Nn, int cshift) {
    int idx = blockIdx.x * blockDim.x + threadIdx.x;
    int i = idx >> cshift;
    int c = idx & ((1 << cshift) - 1);
    if (i >= Nn) return;
    float di = dis[i];
    float y = agg[idx] + h[idx] * di * di + b[c];
    y = fmaxf(y, 0.0f);
    y = (y - m[c]) * (g[c] * rsqrtf(v[c] + EPSV)) + be[c];
    agg[idx] = y;
}

// ---------------- global add pool ----------------
__global__ void gcn_pool(const float* __restrict__ y, const int* __restrict__ batch,
                         float* __restrict__ hg, int Nn, int c4shift, int C) {
    int idx = blockIdx.x * blockDim.x + threadIdx.x;
    int node = idx >> c4shift;
    int cg   = idx & ((1 << c4shift) - 1);
    if (node >= Nn) return;
    int b = batch[node];
    const float4 val = *reinterpret_cast<const float4*>(y + (size_t)node * C + cg * 4);
    float* o = hg + (size_t)b * C + cg * 4;
    atomicAdd(o + 0, val.x);
    atomicAdd(o + 1, val.y);
    atomicAdd(o + 2, val.z);
    atomicAdd(o + 3, val.w);
}

// ---------------- final 64->1 head ----------------
__global__ void gcn_final(const float* __restrict__ hg2, const float* __restrict__ Wf3,
                          const float* __restrict__ bf3, float* __restrict__ out,
                          int G, int K) {
    int g = blockIdx.x * blockDim.x + threadIdx.x;
    if (g >= G) return;
    float s = bf3[0];
    #pragma unroll 8
    for (int k = 0; k < K; ++k) s += hg2[(size_t)g * K + k] * Wf3[k];
    out[g] = fmaxf(s, 0.0f);
}

extern "C" void kernel_launch(void* const* d_in, const int* in_sizes, int n_in,
                              void* d_out, int out_size, void* d_ws, size_t ws_size,
                              hipStream_t stream) {
    (void)in_sizes; (void)n_in; (void)out_size; (void)ws_size;
    const float* x     = (const float*)d_in[0];
    const int*   ei    = (const int*)d_in[1];
    const int*   batch = (const int*)d_in[2];
    const float* W1 = (const float*)d_in[3];  const float* b1 = (const float*)d_in[4];
    const float* g1 = (const float*)d_in[5];  const float* be1= (const float*)d_in[6];
    const float* m1 = (const float*)d_in[7];  const float* v1 = (const float*)d_in[8];
    const float* W2 = (const float*)d_in[9];  const float* b2 = (const float*)d_in[10];
    const float* g2 = (const float*)d_in[11]; const float* be2= (const float*)d_in[12];
    const float* m2 = (const float*)d_in[13]; const float* v2 = (const float*)d_in[14];
    const float* W3 = (const float*)d_in[15]; const float* b3 = (const float*)d_in[16];
    const float* g3 = (const float*)d_in[17]; const float* be3= (const float*)d_in[18];
    const float* m3 = (const float*)d_in[19]; const float* v3 = (const float*)d_in[20];
    const float* Wf1= (const float*)d_in[21]; const float* bf1= (const float*)d_in[22];
    const float* g4 = (const float*)d_in[23]; const float* be4= (const float*)d_in[24];
    const float* m4 = (const float*)d_in[25]; const float* v4 = (const float*)d_in[26];
    const float* Wf2= (const float*)d_in[27]; const float* bf2= (const float*)d_in[28];
    const float* Wf3= (const float*)d_in[29]; const float* bf3= (const float*)d_in[30];

    // workspace layout (floats)
    float* ws  = (float*)d_ws;
    float* dis = ws;                               // 50176 (padded)
    float* P0  = ws + 50176;                       // 50000*256 = 12.8M  (h)
    float* P1  = P0 + 12800000;                    // 50000*256 = 12.8M  (agg / y, in place)
    float* hg  = P1 + 12800000;                    // 1024*128
    float* hg1 = hg + 131072;                      // 1024*64
    float* hg2 = hg1 + 65536;                      // 1024*64

    const dim3 blk(256);
    auto gemm = [&](const float* A, const float* W, float* C, int M, int K, int N,
                    const float* bias, const float* gg, const float* bb,
                    const float* mm, const float* vv, int mode) {
        dim3 grid((N + 63) / 64, (M + 63) / 64);
        gcn_wmma_gemm<<<grid, blk, 0, stream>>>(A, W, C, M, K, N, bias, gg, bb, mm, vv, mode);
    };

    // --- degrees: deg = 1 + count(dst); dis = rsqrt(deg) ---
    gcn_init_deg<<<(NN + 255) / 256, blk, 0, stream>>>(dis, NN);
    gcn_deg_count<<<(NE + 255) / 256, blk, 0, stream>>>(ei, dis, NE);
    gcn_rsqrt_inplace<<<(NN + 255) / 256, blk, 0, stream>>>(dis, NN);

    // --- layer 1: 373 -> 256 ---
    gemm(x, W1, P0, NN, 373, 256, nullptr, nullptr, nullptr, nullptr, nullptr, 0);
    hipMemsetAsync(P1, 0, (size_t)NN * 256 * sizeof(float), stream);
    gcn_aggregate<<<(NE * 64 + 255) / 256, blk, 0, stream>>>(P0, ei, dis, P1, 6, NE, 256);
    gcn_epilogue<<<(NN * 256 + 255) / 256, blk, 0, stream>>>(P1, P0, dis, b1, g1, be1, m1, v1, NN, 8);

    // --- layer 2: 256 -> 128 ---
    gemm(P1, W2, P0, NN, 256, 128, nullptr, nullptr, nullptr, nullptr, nullptr, 0);
    hipMemsetAsync(P1, 0, (size_t)NN * 128 * sizeof(float), stream);
    gcn_aggregate<<<(NE * 32 + 255) / 256, blk, 0, stream>>>(P0, ei, dis, P1, 5, NE, 128);
    gcn_epilogue<<<(NN * 128 + 255) / 256, blk, 0, stream>>>(P1, P0, dis, b2, g2, be2, m2, v2, NN, 7);

    // --- layer 3: 128 -> 128 ---
    gemm(P1, W3, P0, NN, 128, 128, nullptr, nullptr, nullptr, nullptr, nullptr, 0);
    hipMemsetAsync(P1, 0, (size_t)NN * 128 * sizeof(float), stream);
    gcn_aggregate<<<(NE * 32 + 255) / 256, blk, 0, stream>>>(P0, ei, dis, P1, 5, NE, 128);
    gcn_epilogue<<<(NN * 128 + 255) / 256, blk, 0, stream>>>(P1, P0, dis, b3, g3, be3, m3, v3, NN, 7);

    // --- global add pool ---
    hipMemsetAsync(hg, 0, (size_t)NG * 128 * sizeof(float), stream);
    gcn_pool<<<(NN * 32 + 255) / 256, blk, 0, stream>>>(P1, batch, hg, NN, 5, 128);

    // --- MLP head ---
    gemm(hg,  Wf1, hg1, NG, 128, 64, bf1, g4, be4, m4, v4, 2);   // relu + bn4
    gemm(hg1, Wf2, hg2, NG,  64, 64, bf2, nullptr, nullptr, nullptr, nullptr, 1); // relu
    gcn_final<<<(NG + 255) / 256, blk, 0, stream>>>(hg2, Wf3, bf3, (float*)d_out, NG, 64);
}